// GeneralEdgeHeteConv_43903155699827
// MI455X (gfx1250) — compile-verified
//
#include <hip/hip_runtime.h>

typedef __attribute__((ext_vector_type(2))) float v2f;
typedef __attribute__((ext_vector_type(4))) float v4f;
typedef __attribute__((ext_vector_type(8))) float v8f;

// Column-major LDS weight strides (dwords), chosen for zero LDS bank conflicts
// on the 32-lane ds_load_b64 B-fragment fetch:
//  edge: 100 (100%64==36, gcd(36,64)=4 -> 16 consecutive cols hit 16 distinct
//        banks; half-waves differ by k+2 (mod 4) -> disjoint; 64 banks, 1 each)
//  node: 68  (68%64==4, same structure)
#define ESTRIDE 100
#define NSTRIDE 68

__device__ __forceinline__ v8f wmma_f32(v2f a, v2f b, v8f c) {
  // D = A(16x4 f32) * B(4x16 f32) + C(16x16 f32): exact fp32 per reference
  return __builtin_amdgcn_wmma_f32_16x16x4_f32(false, a, false, b, (short)0, c,
                                               false, false);
}

// ---------------------------------------------------------------------------
// out := 0
// ---------------------------------------------------------------------------
__global__ __launch_bounds__(256) void zero_kernel(float* __restrict__ p, int n) {
  int n4 = n >> 2;
  int i = blockIdx.x * 256 + threadIdx.x;
  if (i < n4) ((v4f*)p)[i] = v4f{0.f, 0.f, 0.f, 0.f};
  if (i == 0)
    for (int t = n4 << 2; t < n; ++t) p[t] = 0.f;
}

// ---------------------------------------------------------------------------
// perm := -1, tile_type := 0, counters := 0
// ---------------------------------------------------------------------------
__global__ __launch_bounds__(256) void init_misc_kernel(int* __restrict__ perm,
                                                        int n_perm,
                                                        int* __restrict__ ttype,
                                                        int n_tiles,
                                                        int* __restrict__ ctr) {
  int i = blockIdx.x * 256 + threadIdx.x;
  if (i < n_perm) perm[i] = -1;
  if (i < n_tiles) ttype[i] = 0;
  if (i < 12) ctr[i] = 0;
}

// ---------------------------------------------------------------------------
// Per-type edge counts: ballot -> LDS -> 3 global atomics per workgroup
// (avoids 1.6M serialized atomics on 3 cache lines)
// ---------------------------------------------------------------------------
__global__ __launch_bounds__(256) void count_kernel(const int* __restrict__ etype,
                                                    int n_edges,
                                                    int* __restrict__ ctr) {
  __shared__ int wg[3];
  const int tid = threadIdx.x;
  if (tid < 3) wg[tid] = 0;
  __syncthreads();
  const int i = blockIdx.x * 256 + tid;
  const int t = (i < n_edges) ? etype[i] : -1;
  const int lane = tid & 31;
#pragma unroll
  for (int tt = 0; tt < 3; ++tt) {
    unsigned mask = (unsigned)__ballot(t == tt);
    int cnt = __popc(mask);
    if (lane == 0 && cnt) atomicAdd(&wg[tt], cnt);
  }
  __syncthreads();
  if (tid < 3 && wg[tid]) atomicAdd(&ctr[tid], wg[tid]);
}

// ---------------------------------------------------------------------------
// Bucket bases (16-aligned so every 16-edge tile is single-type); reset cursors
// ctr layout: [0..2]=counts [3..5]=bases [6..8]=cursors
// ---------------------------------------------------------------------------
__global__ void base_kernel(int* __restrict__ ctr) {
  if (threadIdx.x == 0 && blockIdx.x == 0) {
    int c0 = ctr[0], c1 = ctr[1];
    int b1 = (c0 + 15) & ~15;
    int b2 = b1 + ((c1 + 15) & ~15);
    ctr[3] = 0;
    ctr[4] = b1;
    ctr[5] = b2;
    ctr[6] = 0;
    ctr[7] = 0;
    ctr[8] = 0;
  }
}

// ---------------------------------------------------------------------------
// Scatter edge ids into type buckets: lane prefix via ballot, wave offset via
// LDS atomics, workgroup offset via 3 global atomics. Also records tile type.
// ---------------------------------------------------------------------------
__global__ __launch_bounds__(256) void scatter_kernel(
    const int* __restrict__ etype, int n_edges, int* __restrict__ ctr,
    int* __restrict__ perm, int* __restrict__ ttype) {
  __shared__ int wg_cnt[3];
  __shared__ int wg_base[3];
  __shared__ int wv_off[8][3];
  const int tid = threadIdx.x;
  if (tid < 3) wg_cnt[tid] = 0;
  __syncthreads();
  const int i = blockIdx.x * 256 + tid;
  const int t = (i < n_edges) ? etype[i] : -1;
  const int wave = tid >> 5, lane = tid & 31;
  const unsigned low = (lane == 0) ? 0u : (~0u >> (32 - lane));
  int my_pre = 0;
#pragma unroll
  for (int tt = 0; tt < 3; ++tt) {
    unsigned mask = (unsigned)__ballot(t == tt);
    if (t == tt) my_pre = __popc(mask & low);
    if (lane == 0) wv_off[wave][tt] = atomicAdd(&wg_cnt[tt], __popc(mask));
  }
  __syncthreads();
  if (tid < 3) wg_base[tid] = wg_cnt[tid] ? atomicAdd(&ctr[6 + tid], wg_cnt[tid]) : 0;
  __syncthreads();
  if (t >= 0) {
    int slot = ctr[3 + t] + wg_base[t] + wv_off[wave][t] + my_pre;
    perm[slot] = i;
    ttype[slot >> 4] = t;  // all writers of a tile agree (single-type tiles)
  }
}

// ---------------------------------------------------------------------------
// x_out[n] = node_feature[n] @ W_node[node_type[n]]  (64 -> 64)
// One wave per 16-node tile; 2-type A-masking (node kernel is ~6% of the work).
// Weights column-major in LDS -> B-fragment = single conflict-free ds_load_b64.
// ---------------------------------------------------------------------------
__global__ __launch_bounds__(256) void node_xform_kernel(
    const float* __restrict__ x, const float* __restrict__ Wn,
    const int* __restrict__ ntype, float* __restrict__ xout, int n_nodes,
    int n_tiles) {
  __shared__ float w[2 * 64 * NSTRIDE];  // 34 KB
  for (int i = threadIdx.x; i < 2 * 64 * 64; i += 256) {
    int t = i >> 12, k = (i >> 6) & 63, c = i & 63;
    w[(t * 64 + c) * NSTRIDE + k] = Wn[i];
  }
  __syncthreads();

  const int wave = threadIdx.x >> 5;
  const int lane = threadIdx.x & 31;
  const int tile = blockIdx.x * 8 + wave;
  if (tile >= n_tiles) return;  // uniform: EXEC stays all-ones for WMMA

  const int arow = lane & 15;
  const int khalf = (lane >> 4) << 1;
  int node = tile * 16 + arow;
  const bool valid = node < n_nodes;
  if (!valid) node = n_nodes - 1;
  const int t_row = valid ? ntype[node] : -1;  // -1: contributes to no type
  const float* __restrict__ ar = x + (size_t)node * 64;
  const float* __restrict__ wb0 = &w[arow * NSTRIDE];
  const float* __restrict__ wb1 = &w[(64 + arow) * NSTRIDE];

  v8f acc[4] = {};
#pragma unroll
  for (int kk = 0; kk < 64; kk += 4) {
    const int k = kk + khalf;
    v2f a = __builtin_nontemporal_load((const v2f*)(ar + k));  // streamed once
    const v2f z = {0.f, 0.f};
    v2f a0 = (t_row == 0) ? a : z;
    v2f a1 = (t_row == 1) ? a : z;
#pragma unroll
    for (int nb = 0; nb < 4; ++nb) {
      v2f b0 = *(const v2f*)(wb0 + nb * 16 * NSTRIDE + k);
      v2f b1 = *(const v2f*)(wb1 + nb * 16 * NSTRIDE + k);
      acc[nb] = wmma_f32(a0, b0, acc[nb]);
      acc[nb] = wmma_f32(a1, b1, acc[nb]);
    }
  }

  const int rbase = (lane >> 4) * 8;
  const int col = lane & 15;
#pragma unroll
  for (int r = 0; r < 8; ++r) {
    const int orow = tile * 16 + rbase + r;
    if (orow < n_nodes) {
#pragma unroll
      for (int nb = 0; nb < 4; ++nb)
        xout[(size_t)orow * 64 + nb * 16 + col] = ((const float*)&acc[nb])[r];
    }
  }
}

// ---------------------------------------------------------------------------
// msg[e] = concat(x_out[src[e]], ef[e]) @ W_msg[type]; out[dst[e]] += msg
// Tiles are single-type via perm[] -> exactly one GEMM per tile (96 WMMA).
// Padded rows redirect to edge 0 (finite data); since D row M depends only on
// A row M, their D rows are simply not scattered -> no per-load masking VALU.
// K split at 64: x_out gather (L2-resident) regular, edge_feature stream NT.
// Scatter metadata (perm/dst) moved across lanes with shuffles, not re-loads.
// ---------------------------------------------------------------------------
__global__ __launch_bounds__(256) void edge_msg_kernel(
    const float* __restrict__ xout, const float* __restrict__ ef,
    const float* __restrict__ Wm, const int* __restrict__ perm,
    const int* __restrict__ ttype, const int* __restrict__ src,
    const int* __restrict__ dst, float* __restrict__ out, int n_tiles) {
  __shared__ float w[3 * 64 * ESTRIDE];  // 75 KB
  for (int i = threadIdx.x; i < 3 * 96 * 64; i += 256) {
    int t = i / (96 * 64);
    int rem = i - t * 96 * 64;
    int k = rem >> 6, c = rem & 63;
    w[(t * 64 + c) * ESTRIDE + k] = Wm[i];
  }
  __syncthreads();

  const int wave = threadIdx.x >> 5;
  const int lane = threadIdx.x & 31;
  const int tile = blockIdx.x * 8 + wave;
  if (tile >= n_tiles) return;  // uniform

  const int arow = lane & 15;
  const int khalf = (lane >> 4) << 1;
  const int e = perm[tile * 16 + arow];  // lanes 0-15 hold rows 0-15
  const int es = (e >= 0) ? e : 0;       // padded rows read edge 0 (discarded)
  const int t = ttype[tile];             // single type per tile by construction
  const int s = src[es];
  const int dd = dst[es];
  const float* __restrict__ xr = xout + (size_t)s * 64;
  const float* __restrict__ er = ef + (size_t)es * 32;
  const float* __restrict__ wb = &w[(t * 64 + arow) * ESTRIDE];

  v8f acc[4] = {};
  // K = 0..63: gathered x_out rows (hot in L2: 25.6 MB << 192 MB)
#pragma unroll
  for (int kk = 0; kk < 64; kk += 4) {
    const int k = kk + khalf;
    v2f a = *(const v2f*)(xr + k);
#pragma unroll
    for (int nb = 0; nb < 4; ++nb) {
      v2f b = *(const v2f*)(wb + nb * 16 * ESTRIDE + k);
      acc[nb] = wmma_f32(a, b, acc[nb]);
    }
  }
  // K = 64..95: edge_feature, 205 MB stream -> non-temporal (protect L2)
#pragma unroll
  for (int kk = 0; kk < 32; kk += 4) {
    const int k = kk + khalf;
    v2f a = __builtin_nontemporal_load((const v2f*)(er + k));
#pragma unroll
    for (int nb = 0; nb < 4; ++nb) {
      v2f b = *(const v2f*)(wb + nb * 16 * ESTRIDE + 64 + k);
      acc[nb] = wmma_f32(a, b, acc[nb]);
    }
  }

  // C/D layout: VGPR r, lanes 0-15 -> M=r, lanes 16-31 -> M=8+r; col = lane%16
  const int rbase = (lane >> 4) * 8;
  const int col = lane & 15;
#pragma unroll
  for (int r = 0; r < 8; ++r) {
    const int eo = __shfl(e, rbase + r, 32);
    const int dr = __shfl(dd, rbase + r, 32);
    if (eo >= 0) {
      float* orow = out + (size_t)dr * 64 + col;
#pragma unroll
      for (int nb = 0; nb < 4; ++nb)
        unsafeAtomicAdd(orow + nb * 16, ((const float*)&acc[nb])[r]);
    }
  }
}

// ---------------------------------------------------------------------------
extern "C" void kernel_launch(void* const* d_in, const int* in_sizes, int n_in,
                              void* d_out, int out_size, void* d_ws,
                              size_t ws_size, hipStream_t stream) {
  const float* node_feature = (const float*)d_in[0];
  const float* edge_feature = (const float*)d_in[1];
  const float* W_node = (const float*)d_in[2];
  const float* W_msg = (const float*)d_in[3];
  const int* node_type = (const int*)d_in[4];
  const int* edge_type = (const int*)d_in[5];
  const int* edge_index = (const int*)d_in[6];
  const int n_nodes = in_sizes[4];
  const int n_edges = in_sizes[5];
  const int* src = edge_index;
  const int* dst = edge_index + n_edges;

  float* out = (float*)d_out;

  // Workspace layout: [x_out][perm][tile_type][counters]
  float* xout = (float*)d_ws;
  const int edge_tiles = (n_edges + 48) / 16 + 1;  // covers per-bucket padding
  const int n_perm = edge_tiles * 16;
  int* perm = (int*)((char*)d_ws + (size_t)n_nodes * 64 * sizeof(float));
  int* ttype = perm + n_perm;
  int* ctr = ttype + edge_tiles;

  zero_kernel<<<((out_size >> 2) + 255) / 256, 256, 0, stream>>>(out, out_size);
  init_misc_kernel<<<(n_perm + 255) / 256, 256, 0, stream>>>(perm, n_perm,
                                                             ttype, edge_tiles,
                                                             ctr);
  count_kernel<<<(n_edges + 255) / 256, 256, 0, stream>>>(edge_type, n_edges,
                                                          ctr);
  base_kernel<<<1, 32, 0, stream>>>(ctr);
  scatter_kernel<<<(n_edges + 255) / 256, 256, 0, stream>>>(edge_type, n_edges,
                                                            ctr, perm, ttype);

  const int node_tiles = (n_nodes + 15) / 16;
  node_xform_kernel<<<(node_tiles + 7) / 8, 256, 0, stream>>>(
      node_feature, W_node, node_type, xout, n_nodes, node_tiles);

  edge_msg_kernel<<<(edge_tiles + 7) / 8, 256, 0, stream>>>(
      xout, edge_feature, W_msg, perm, ttype, src, dst, out, edge_tiles);
}